// FCM_21560735826243
// MI455X (gfx1250) — compile-verified
//
#include <hip/hip_runtime.h>

// ---------------------------------------------------------------------------
// Fuzzy C-Means membership, fused:  u = rownorm( cdist(x,c)^(-2/(m-1)) )
// x: (N=262144, F=128) f32   centers: (K=256, F=128) f32   out: (N, 256) f32
//
// One pass: f32 WMMA GEMM (x @ c^T) + distance + power law + row-normalize.
// Wave32: each wave computes a 16-row x 256-col slab (16 WMMA C-tiles held in
// 128 VGPRs); row sums reduce with shfl_xor inside 16-lane halves.
// Staging uses CDNA5 async global->LDS DMA (ASYNCcnt) instead of VGPR bounce.
// ---------------------------------------------------------------------------

typedef float v2f __attribute__((ext_vector_type(2)));
typedef float v8f __attribute__((ext_vector_type(8)));

#define K_CL          256
#define F_DIM         128
#define LDS_STRIDE    132   // padded row stride in floats (128 + 4) -> bank-conflict-free frags
#define LDS_STRIDE4   33    // in float4
#define ROWS_PER_WAVE 16
#define WAVES_PER_WG  8
#define ROWS_PER_WG   128
#define NTILES        16    // K_CL / 16
// m = 1.7 ; u = d^(-2/(m-1)) = (d^2)^(-1/(m-1)) ; clamp d>=1e-12 <=> d2>=1e-24
#define POW_EXP_D2    (-1.4285714285714286f)

// LDS byte-offset map (dynamic LDS base = 0: no static __shared__ declared)
#define SC_OFF_BYTES  0u
#define SX_OFF_BYTES  ((unsigned)(K_CL * LDS_STRIDE * 4))

// Async DMA: global (SADDR base + 32-bit VGPR offset) -> LDS (VGPR byte offset).
// GVS form, 16 bytes per lane. Tracked by ASYNCcnt.
__device__ __forceinline__ void async_g2l_b128(unsigned lds_byte_off,
                                               unsigned gmem_byte_off,
                                               const void* gmem_base) {
  asm volatile("global_load_async_to_lds_b128 %0, %1, %2"
               :: "v"(lds_byte_off), "v"(gmem_byte_off), "s"(gmem_base)
               : "memory");
}

__global__ __launch_bounds__(256) void fcm_wmma_kernel(
    const float* __restrict__ x, const float* __restrict__ c,
    float* __restrict__ out)
{
  extern __shared__ float smem[];
  float* sC  = smem;                           // 256 * 132 floats (padded centers)
  float* sX  = sC + K_CL * LDS_STRIDE;         // 128 * 132 floats (padded x tile)
  float* sC2 = sX + ROWS_PER_WG * LDS_STRIDE;  // 256 floats  ||c||^2

  const int tid  = threadIdx.x;
  const int lane = tid & 31;
  const int wave = tid >> 5;
  const int nl   = lane & 15;   // N (col) index inside tile / row index for frags
  const int h    = lane >> 4;   // lane half

  const int waveRow0 = blockIdx.x * ROWS_PER_WG + wave * ROWS_PER_WAVE;

  // ---- Stage centers into LDS: async DMA, all 256 threads, 16B/lane ----
  {
    #pragma unroll
    for (int i = 0; i < (K_CL * F_DIM / 4) / 256; ++i) {   // 32 iterations
      unsigned idx = (unsigned)(i * 256 + tid);            // float4 index
      unsigned row = idx >> 5;                             // 32 float4 per row
      unsigned col = idx & 31;
      async_g2l_b128(SC_OFF_BYTES + (row * LDS_STRIDE4 + col) * 16u,
                     idx * 16u, (const void*)c);
    }
  }
  // ---- Stage this wave's 16 x-rows into LDS: async DMA ----
  {
    const void* xbase = (const void*)(x + (size_t)waveRow0 * F_DIM);
    const unsigned sxWave = SX_OFF_BYTES +
        (unsigned)(wave * ROWS_PER_WAVE * LDS_STRIDE * 4);
    #pragma unroll
    for (int i = 0; i < (ROWS_PER_WAVE * F_DIM / 4) / 32; ++i) {  // 16 iterations
      unsigned idx = (unsigned)(i * 32 + lane);
      unsigned row = idx >> 5;
      unsigned col = idx & 31;
      async_g2l_b128(sxWave + (row * LDS_STRIDE4 + col) * 16u,
                     idx * 16u, xbase);
    }
  }
  asm volatile("s_wait_asynccnt 0" ::: "memory");  // DMA done before anyone reads LDS
  __syncthreads();

  // ---- ||c_t||^2 : one thread per center, rotated start to spread banks ----
  {
    const float4* r4 = (const float4*)sC + tid * LDS_STRIDE4;
    float s = 0.f;
    #pragma unroll
    for (int i = 0; i < 32; ++i) {
      float4 v = r4[(tid + i) & 31];
      s += v.x * v.x + v.y * v.y + v.z * v.z + v.w * v.w;
    }
    sC2[tid] = s;
  }
  __syncthreads();

  // ---- ||x_row||^2 for this wave's rows; distribute to C-tile row slots ----
  float x2r[8];
  {
    const float4* r4 = (const float4*)sX + (wave * ROWS_PER_WAVE + nl) * LDS_STRIDE4;
    float p = 0.f;
    #pragma unroll
    for (int i = 0; i < 16; ++i) {            // half-row (64 floats) per lane
      float4 v = r4[h * 16 + ((nl + i) & 15)];
      p += v.x * v.x + v.y * v.y + v.z * v.z + v.w * v.w;
    }
    p += __shfl_xor(p, 16, 32);               // combine halves: lane m & m+16 -> x2[row m]
    #pragma unroll
    for (int r = 0; r < 8; ++r)               // C-tile element (r, lane) is row r + 8*h
      x2r[r] = __shfl(p, r + 8 * h, 32);
  }

  // ---- Main GEMM: 16 accumulator tiles, F/4 = 32 WMMA k-steps each ----
  v8f acc[NTILES];
  #pragma unroll
  for (int t = 0; t < NTILES; ++t) {
    v8f z = {0.f, 0.f, 0.f, 0.f, 0.f, 0.f, 0.f, 0.f};
    acc[t] = z;
  }

  // A frag (16x4 f32): lane(m=nl, half h) holds x[m][k+2h], x[m][k+2h+1]
  const float* aPtr = sX + (wave * ROWS_PER_WAVE + nl) * LDS_STRIDE + 2 * h;
  // B frag (4x16 f32): lane(n=nl, half h) holds c[n][k+2h], c[n][k+2h+1]
  // Hoist all 16 tile base pointers -> every ds_load is base + imm(k*4).
  const float* bP[NTILES];
  #pragma unroll
  for (int t = 0; t < NTILES; ++t)
    bP[t] = sC + (t * 16 + nl) * LDS_STRIDE + 2 * h;

  #pragma unroll 4
  for (int k = 0; k < F_DIM; k += 4) {
    v2f a = *(const v2f*)(aPtr + k);
    #pragma unroll
    for (int t = 0; t < NTILES; ++t) {
      v2f b = *(const v2f*)(bP[t] + k);
      acc[t] = __builtin_amdgcn_wmma_f32_16x16x4_f32(
          /*neg_a=*/false, a, /*neg_b=*/false, b,
          /*c_mod=*/(short)0, acc[t], /*reuse_a=*/false, /*reuse_b=*/false);
    }
  }

  // ---- Epilogue: d^2 -> power law, row-sum, normalize, store ----
  float rs[8];
  #pragma unroll
  for (int r = 0; r < 8; ++r) rs[r] = 0.f;

  #pragma unroll
  for (int t = 0; t < NTILES; ++t) {
    float c2v = sC2[t * 16 + nl];
    #pragma unroll
    for (int r = 0; r < 8; ++r) {
      float d2 = x2r[r] + c2v - 2.0f * acc[t][r];
      d2 = fmaxf(d2, 1e-24f);                       // == max(d,1e-12) on d
      // u = d2^(-1/(m-1)) = exp2( -1.42857.. * log2(d2) )
      // v_log_f32 / v_exp_f32 (TRANS pipe, co-executes w/ XDL)
      float u = __builtin_amdgcn_exp2f(POW_EXP_D2 * __builtin_amdgcn_logf(d2));
      acc[t][r] = u;
      rs[r] += u;
    }
  }

  // Row sum lives across the 16 lanes of each half (N dim): butterfly within half.
  #pragma unroll
  for (int r = 0; r < 8; ++r) {
    #pragma unroll
    for (int s = 1; s < 16; s <<= 1) rs[r] += __shfl_xor(rs[r], s, 32);
    rs[r] = 1.0f / rs[r];
  }

  #pragma unroll
  for (int t = 0; t < NTILES; ++t) {
    #pragma unroll
    for (int r = 0; r < 8; ++r) {
      int grow = waveRow0 + r + 8 * h;
      out[(size_t)grow * K_CL + t * 16 + nl] = acc[t][r] * rs[r];
    }
  }
}

extern "C" void kernel_launch(void* const* d_in, const int* in_sizes, int n_in,
                              void* d_out, int out_size, void* d_ws, size_t ws_size,
                              hipStream_t stream) {
  const float* x = (const float*)d_in[0];     // (N, 128)
  const float* c = (const float*)d_in[1];     // (256, 128)
  float* out = (float*)d_out;                 // (N, 256)

  const int N = in_sizes[0] / F_DIM;          // 262144
  const int nblocks = N / ROWS_PER_WG;        // 2048

  const size_t smem_bytes =
      (size_t)(K_CL * LDS_STRIDE + ROWS_PER_WG * LDS_STRIDE + K_CL) * sizeof(float);
  // ~199 KB of the 320 KB WGP LDS

  fcm_wmma_kernel<<<nblocks, 256, smem_bytes, stream>>>(x, c, out);
}